// GATAttention_56444460204099
// MI455X (gfx1250) — compile-verified
//
#include <hip/hip_runtime.h>

// GAT attention layer for MI455X (gfx1250, wave32).
// Shapes fixed by the reference: BS=8, N=2048, F=128.
// Memory-bound on att_mat (134 MB x 2 passes => ~11.5 us @ 23.3 TB/s), so both
// GEMMs use full-precision V_WMMA_F32_16X16X4_F32 (fp32 fidelity is free here).
// CDNA5 paths: v_wmma f32, TENSOR_LOAD_TO_LDS (TDM) staging, s_wait_tensorcnt,
// global_prefetch_b8 on the HBM-bound att stream, conflict-free padded LDS.

typedef __attribute__((ext_vector_type(2))) float v2f;
typedef __attribute__((ext_vector_type(8))) float v8f;
typedef __attribute__((ext_vector_type(4))) unsigned int u32x4;
typedef __attribute__((ext_vector_type(8))) int i32x8;
typedef __attribute__((ext_vector_type(4))) int i32x4;

constexpr int BS = 8;
constexpr int N  = 2048;
constexpr int F  = 128;
constexpr int LDW = 132;   // padded LDS row stride for W^T  (132 % 64 == 4 -> conflict-free)
constexpr int LDA = 2052;  // padded LDS row stride for attn weights (2052 % 64 == 4)

__device__ __forceinline__ float wave_max(float v) {
#pragma unroll
  for (int o = 16; o > 0; o >>= 1) v = fmaxf(v, __shfl_xor(v, o, 32));
  return v;
}
__device__ __forceinline__ float wave_sum(float v) {
#pragma unroll
  for (int o = 16; o > 0; o >>= 1) v += __shfl_xor(v, o, 32);
  return v;
}

// ---------------------------------------------------------------------------
// TDM: issue TENSOR_LOAD_TO_LDS of a 2D fp32 tile (clang-23 6-arg builtin:
// g0 u32x4, g1 i32x8, g2 i32x4, g3 i32x4, i32x8, cpol).
// Descriptor per CDNA5 ISA ch.8: group0 = {count/lds_addr/global_addr/type},
// group1 = {flags, tensor dims, tile dims, dim0 stride}.
// ---------------------------------------------------------------------------
__device__ __forceinline__ void tdm_load_f32_2d(unsigned ldsOff, const void* gsrc,
                                                unsigned dim0, unsigned dim1,
                                                unsigned tile0, unsigned tile1,
                                                unsigned stride0) {
  const unsigned long long ga = (unsigned long long)(size_t)gsrc;
  u32x4 g0;
  g0[0] = 1u;                                              // count=1 (valid), user mode
  g0[1] = ldsOff;                                          // lds_addr (bytes)
  g0[2] = (unsigned)ga;                                    // global_addr[31:0]
  g0[3] = ((unsigned)(ga >> 32) & 0x01FFFFFFu) | (2u << 30);  // addr[56:32] | type=2
  i32x8 g1;
  g1[0] = (int)(2u << 16);                                 // data_size=4B
  g1[1] = (int)(dim0 << 16);                               // tensor_dim0[15:0] @ bits63:48
  g1[2] = (int)((dim0 >> 16) | (dim1 << 16));              // dim0 hi | tensor_dim1 lo
  g1[3] = (int)((dim1 >> 16) | (tile0 << 16));             // dim1 hi | tile_dim0
  g1[4] = (int)(tile1 & 0xFFFFu);                          // tile_dim1 (tile_dim2=0)
  g1[5] = (int)stride0;                                    // tensor_dim0_stride lo
  g1[6] = 0;
  g1[7] = 0;
  i32x4 gz;
  gz[0] = 0; gz[1] = 0; gz[2] = 0; gz[3] = 0;              // groups 2/3 unused (<=2D)
  i32x8 gz8;
  gz8[0] = 0; gz8[1] = 0; gz8[2] = 0; gz8[3] = 0;
  gz8[4] = 0; gz8[5] = 0; gz8[6] = 0; gz8[7] = 0;
  __builtin_amdgcn_tensor_load_to_lds(g0, g1, gz, gz, gz8, 0);
}

// ---------------------------------------------------------------------------
// Kernel A: H = X @ W  (flattened [BS*N, F] x [F, F]).
// 256 threads = 8 waves; each wave produces a 16x128 tile.
// W is TDM-loaded raw into LDS, then transposed once into Wt[n][k] with padded
// stride 132 so every B-fragment (W[k][n], W[k+1][n]) is one contiguous
// 8B-aligned ds_load_b64 into an even VGPR pair (no repack movs, and the b64
// reads cover all 64 banks conflict-free).
// ---------------------------------------------------------------------------
__global__ __launch_bounds__(256)
void gat_h_gemm(const float* __restrict__ X, const float* __restrict__ W,
                float* __restrict__ H) {
  extern __shared__ float lds[];
  float* wraw = lds;           // F*F floats  : TDM destination (row-major W)
  float* wt   = lds + F * F;   // F*LDW floats: transposed, padded Wt[n][k]

  const int tid  = threadIdx.x;
  const int lane = tid & 31;
  const int wave = tid >> 5;

  if (tid < 32) {  // one wave issues the DMA (TDM ignores EXEC; wave-level op)
    tdm_load_f32_2d((unsigned)(size_t)(void*)wraw, W, F, F, F, F, F);
    __builtin_amdgcn_s_wait_tensorcnt(0);
  }
  __syncthreads();

  // one-time transpose into padded layout
  for (int idx = tid; idx < F * F; idx += 256) {
    const int k = idx >> 7, n = idx & (F - 1);
    wt[n * LDW + k] = wraw[idx];
  }
  __syncthreads();

  const int rowBase = blockIdx.x * 128 + wave * 16;
  const int mrow    = lane & 15;        // A-matrix row / D-matrix column lane
  const int khalf   = (lane >> 4) * 2;  // lane-half selects K, K+2
  const float* Xrow = X + (size_t)(rowBase + mrow) * F;

  v8f acc[8] = {};
#pragma unroll 4
  for (int k0 = 0; k0 < F; k0 += 4) {
    const v2f a = *(const v2f*)(Xrow + k0 + khalf);   // global_load_b64
#pragma unroll
    for (int jt = 0; jt < 8; ++jt) {
      const v2f bf = *(const v2f*)(wt + (jt * 16 + mrow) * LDW + k0 + khalf);  // ds_load_b64
      acc[jt] = __builtin_amdgcn_wmma_f32_16x16x4_f32(
          false, a, false, bf, (short)0, acc[jt], false, false);
    }
  }

#pragma unroll
  for (int jt = 0; jt < 8; ++jt) {
#pragma unroll
    for (int v = 0; v < 8; ++v) {
      const int m = v + ((lane >> 4) << 3);  // C/D layout: M = v + 8*(lane>=16)
      H[(size_t)(rowBase + m) * F + jt * 16 + mrow] = acc[jt][v];
    }
  }
}

// ---------------------------------------------------------------------------
// Kernel B: s1 = H . a[:F], s2 = H . a[F:]  (one row per thread; tiny)
// ---------------------------------------------------------------------------
__global__ void gat_s1s2(const float* __restrict__ H, const float* __restrict__ a,
                         float* __restrict__ s1, float* __restrict__ s2) {
  const int g = blockIdx.x * blockDim.x + threadIdx.x;  // [0, BS*N)
  const float* hr = H + (size_t)g * F;
  float a1 = 0.f, a2 = 0.f;
#pragma unroll 8
  for (int f = 0; f < F; ++f) {
    const float hv = hr[f];
    a1 += hv * a[f];
    a2 += hv * a[F + f];
  }
  s1[g] = a1;
  s2[g] = a2;
}

// ---------------------------------------------------------------------------
// Kernel C: su[b, j] = sum_i att[b, i, j]  (coalesced across j, prefetched)
// ---------------------------------------------------------------------------
__global__ void gat_colsum(const float* __restrict__ att, float* __restrict__ su) {
  const int j = blockIdx.x * blockDim.x + threadIdx.x;
  const int b = blockIdx.y;
  const float* p = att + (size_t)b * N * N + j;
  float s = 0.f;
  for (int i = 0; i < N; i += 4) {
    __builtin_prefetch(p + (size_t)(i + 16) * N, 0, 1);   // global_prefetch_b8
#pragma unroll
    for (int u = 0; u < 4; ++u) s += p[(size_t)(i + u) * N];
  }
  su[b * N + j] = s;
}

// ---------------------------------------------------------------------------
// Kernel D: masked tanh attention + row softmax + (attn @ H) + su==0 fallback.
// Block = (b, 16-row tile). 8 waves. s2[b,:] (8 KB) staged to LDS via TDM.
// Phase 1 builds the 16x2048 softmax-weight tile in padded LDS; phase 2: each
// wave runs a K=2048 fp32 WMMA chain for one 16-column slice of the output,
// A-frags via ds_load_b64 (conflict-free), B-frags from L2-resident H.
// ---------------------------------------------------------------------------
__global__ __launch_bounds__(256)
void gat_attn(const float* __restrict__ att, const float* __restrict__ H,
              const float* __restrict__ s1, const float* __restrict__ s2,
              const float* __restrict__ su, const float* __restrict__ X,
              float* __restrict__ out) {
  extern __shared__ float lds[];
  float* wtile  = lds;                  // 16 * LDA floats: scores -> weights
  float* rowSum = lds + 16 * LDA;       // 16 floats
  float* s2s    = rowSum + 16;          // N floats (TDM-staged s2[b,:])

  const int tid  = threadIdx.x;
  const int lane = tid & 31;
  const int wave = tid >> 5;
  const int b    = blockIdx.y;
  const int i0   = blockIdx.x * 16;

  if (tid < 32) {  // stage s2[b,:] (2048 fp32, contiguous) via the TDM
    tdm_load_f32_2d((unsigned)(size_t)(void*)s2s, s2 + (size_t)b * N, N, 1, N, 1, N);
    __builtin_amdgcn_s_wait_tensorcnt(0);
  }
  __syncthreads();

  // ---- Phase 1: two rows per wave ----
#pragma unroll
  for (int rr = 0; rr < 2; ++rr) {
    const int r = wave * 2 + rr;
    const float s1r = s1[b * N + i0 + r];
    const float* arow = att + ((size_t)b * N + (i0 + r)) * N;

    float mx = -3.0e15f;
    for (int j = lane; j < N; j += 32) {
      __builtin_prefetch(arow + j + 512, 0, 1);           // global_prefetch_b8
      const float mask = arow[j];
      const float al   = tanhf(s1r + s2s[j]);
      const float val  = (mask > 0.f) ? al : -1.0e15f;    // reference NEG_INF
      wtile[r * LDA + j] = val;
      mx = fmaxf(mx, val);
    }
    mx = wave_max(mx);

    float sum = 0.f;
    for (int j = lane; j < N; j += 32) {
      const float w = __expf(wtile[r * LDA + j] - mx);
      wtile[r * LDA + j] = w;
      sum += w;
    }
    sum = wave_sum(sum);
    if (lane == 0) rowSum[r] = sum;
  }
  __syncthreads();

  // ---- Phase 2: out[i0:i0+16, n0:n0+16] = weights(16x2048) @ H_b(2048, n0:n0+16)
  const int n0    = wave * 16;
  const int mrow  = lane & 15;
  const int khalf = (lane >> 4) * 2;
  const float* Hb = H + (size_t)b * N * F;

  v8f acc = {};
  for (int k0 = 0; k0 < N; k0 += 4) {
    const v2f a = *(const v2f*)(wtile + mrow * LDA + k0 + khalf);  // ds_load_b64
    v2f bf;
    bf.x = Hb[(size_t)(k0 + khalf) * F + n0 + mrow];
    bf.y = Hb[(size_t)(k0 + khalf + 1) * F + n0 + mrow];
    acc = __builtin_amdgcn_wmma_f32_16x16x4_f32(
        false, a, false, bf, (short)0, acc, false, false);
  }

#pragma unroll
  for (int v = 0; v < 8; ++v) {
    const int m    = v + ((lane >> 4) << 3);
    const int grow = i0 + m;
    const float res = acc[v] / rowSum[m];
    const float suv = su[b * N + grow];
    const size_t o  = ((size_t)b * N + grow) * F + n0 + mrow;
    out[o] = (suv == 0.f) ? X[o] : res;
  }
}

// ---------------------------------------------------------------------------
extern "C" void kernel_launch(void* const* d_in, const int* in_sizes, int n_in,
                              void* d_out, int out_size, void* d_ws, size_t ws_size,
                              hipStream_t stream) {
  const float* X   = (const float*)d_in[0];  // [BS, N, F]
  const float* att = (const float*)d_in[1];  // [BS, N, N]
  const float* W   = (const float*)d_in[2];  // [F, F]
  const float* a   = (const float*)d_in[3];  // [2F]
  float* out = (float*)d_out;                // [BS, N, F]

  // Workspace layout (~8.4 MB total)
  float* H  = (float*)d_ws;                  // BS*N*F
  float* s1 = H + (size_t)BS * N * F;        // BS*N
  float* s2 = s1 + BS * N;                   // BS*N
  float* su = s2 + BS * N;                   // BS*N

  const int ldsA = (F * F + F * LDW) * (int)sizeof(float);       // ~130 KB (W + Wt)
  const int ldsC = (16 * LDA + 16 + N) * (int)sizeof(float);     // ~136 KB
  (void)hipFuncSetAttribute((const void*)gat_h_gemm,
                            hipFuncAttributeMaxDynamicSharedMemorySize, ldsA);
  (void)hipFuncSetAttribute((const void*)gat_attn,
                            hipFuncAttributeMaxDynamicSharedMemorySize, ldsC);

  gat_h_gemm<<<dim3((BS * N) / 128), dim3(256), ldsA, stream>>>(X, W, H);
  gat_s1s2 <<<dim3((BS * N) / 256), dim3(256), 0,    stream>>>(H, a, s1, s2);
  gat_colsum<<<dim3(N / 256, BS),   dim3(256), 0,    stream>>>(att, su);
  gat_attn <<<dim3(N / 16, BS),     dim3(256), ldsC, stream>>>(att, H, s1, s2, su, X, out);
}